// BertLexer_90804198572717
// MI455X (gfx1250) — compile-verified
//
#include <hip/hip_runtime.h>
#include <hip/hip_bf16.h>

#define L_ 4
#define B_ 32
#define S_ 512
#define F_ 768
#define W_ 257   // W_MAX + 1

typedef __attribute__((ext_vector_type(2))) float v2f;
typedef __attribute__((ext_vector_type(8))) float v8f;

__device__ __forceinline__ int lower_bound_lds(const int* a, int n, int v) {
    int lo = 0, hi = n;
    while (lo < hi) {
        int mid = (lo + hi) >> 1;
        if (a[mid] < v) lo = mid + 1; else hi = mid;
    }
    return lo;
}

// ---------------------------------------------------------------------------
// Kernel 1: sub[b,s,f] = gamma * sum_l softmax(lw)_l * h[l,b,s,f]
// Pure streaming pass: 201 MB read, 48 MB write (workspace stays L2-resident).
// ---------------------------------------------------------------------------
__global__ void mix_kernel(const float* __restrict__ h,
                           const float* __restrict__ lw,
                           const float* __restrict__ gamma,
                           float* __restrict__ sub) {
    const size_t N = (size_t)B_ * S_ * F_;
    float w0 = lw[0], w1 = lw[1], w2 = lw[2], w3 = lw[3];
    float mx = fmaxf(fmaxf(w0, w1), fmaxf(w2, w3));
    float e0 = __expf(w0 - mx), e1 = __expf(w1 - mx);
    float e2 = __expf(w2 - mx), e3 = __expf(w3 - mx);
    float inv = gamma[0] / (e0 + e1 + e2 + e3);
    e0 *= inv; e1 *= inv; e2 *= inv; e3 *= inv;

    size_t i = ((size_t)blockIdx.x * blockDim.x + threadIdx.x) * 4;
    if (i + 3 < N) {
        float4 a = *(const float4*)(h + 0 * N + i);
        float4 b = *(const float4*)(h + 1 * N + i);
        float4 c = *(const float4*)(h + 2 * N + i);
        float4 d = *(const float4*)(h + 3 * N + i);
        float4 o;
        o.x = e0 * a.x + e1 * b.x + e2 * c.x + e3 * d.x;
        o.y = e0 * a.y + e1 * b.y + e2 * c.y + e3 * d.y;
        o.z = e0 * a.z + e1 * b.z + e2 * c.z + e3 * d.z;
        o.w = e0 * a.w + e1 * b.w + e2 * c.w + e3 * d.w;
        *(float4*)(sub + i) = o;
    }
}

// ---------------------------------------------------------------------------
// Kernel 2: segment means for words 1..256 via V_WMMA_F32_16X16X4_F32.
//   D(16 words x 16 feats) += A(16x4 one-hot indicator) * B(4x16 sub tile)
// One wave per (f-tile, word-tile, batch). word_ids sorted per row => each
// 16-word tile covers one contiguous s-span (binary search on LDS copy).
// Sortedness also means (id == wm) alone is a correct indicator: ids outside
// [lo,hi) can never equal wm, so the A build is branch-free (EXEC stays all-1s
// around the WMMA, as required).
// ---------------------------------------------------------------------------
__global__ void seg_mean_wmma(const float* __restrict__ sub,
                              const int* __restrict__ wid,
                              float* __restrict__ out) {
    const int lane  = threadIdx.x;      // 0..31, wave32
    const int ft    = blockIdx.x;       // 0..47  (F/16)
    const int wt    = blockIdx.y;       // 0..15  (256 words / 16)
    const int b     = blockIdx.z;       // 0..31
    const int wbase = 1 + 16 * wt;      // words 1..256 (slot 0 handled elsewhere)
    const int fbase = 16 * ft;

    __shared__ int   srow[S_];          // 2 KB staged word_ids row
    __shared__ int   bnd[17];           // span boundaries
    __shared__ float wscale[16];        // 1/count (0 for empty words)

    // stage word ids with b128 loads: 4 int4 per lane
    const int4* row4 = (const int4*)(wid + b * S_);
#pragma unroll
    for (int i = 0; i < S_ / 4; i += 32) ((int4*)srow)[i + lane] = row4[i + lane];
    __syncthreads();

    if (lane < 17) bnd[lane] = lower_bound_lds(srow, S_, wbase + lane);
    __syncthreads();
    const int lo = bnd[0], hi = bnd[16];            // uniform across the wave
    if (lane < 16) {
        int cnt = bnd[lane + 1] - bnd[lane];
        wscale[lane] = (cnt > 0) ? 1.0f / (float)cnt : 0.0f;
    }
    __syncthreads();

    const int half = lane >> 4;         // 0: lanes 0-15, 1: lanes 16-31
    const int mA   = lane & 15;         // A-matrix row held by this lane
    const int n    = lane & 15;         // B/D column held by this lane
    const int kA   = half * 2;          // A: K={0,1} (lo half) / K={2,3} (hi half)
    const int kB   = half * 2;          // B: VGPR r -> K=r (+2 for upper half)
    const int wm   = wbase + mA;

    const float* scol = sub + (size_t)b * S_ * F_ + fbase + n;
    const int*   sa   = srow + kA;      // even index => 8B-aligned int2 reads

    v8f acc0 = {0.f,0.f,0.f,0.f,0.f,0.f,0.f,0.f};
    v8f acc1 = {0.f,0.f,0.f,0.f,0.f,0.f,0.f,0.f};

    int s0 = lo & ~3;
    const float* p = scol + (size_t)(s0 + kB) * F_;
    // main loop: two independent WMMA chains; all bounds uniform (scalar branch)
    for (; s0 + 4 < hi; s0 += 8, p += 8 * F_) {
        int2 w0 = *(const int2*)(sa + s0);
        v2f a0; a0.x = (w0.x == wm) ? 1.f : 0.f;
                a0.y = (w0.y == wm) ? 1.f : 0.f;
        v2f b0; b0.x = p[0]; b0.y = p[F_];
        acc0 = __builtin_amdgcn_wmma_f32_16x16x4_f32(
            false, a0, false, b0, (short)0, acc0, false, false);

        int2 w1 = *(const int2*)(sa + s0 + 4);
        v2f a1; a1.x = (w1.x == wm) ? 1.f : 0.f;
                a1.y = (w1.y == wm) ? 1.f : 0.f;
        v2f b1; b1.x = p[4 * F_]; b1.y = p[5 * F_];
        acc1 = __builtin_amdgcn_wmma_f32_16x16x4_f32(
            false, a1, false, b1, (short)0, acc1, false, false);
    }
    if (s0 < hi) {                      // uniform tail: one 4-chunk
        int2 w0 = *(const int2*)(sa + s0);
        v2f a0; a0.x = (w0.x == wm) ? 1.f : 0.f;
                a0.y = (w0.y == wm) ? 1.f : 0.f;
        v2f b0; b0.x = p[0]; b0.y = p[F_];
        acc0 = __builtin_amdgcn_wmma_f32_16x16x4_f32(
            false, a0, false, b0, (short)0, acc0, false, false);
    }

    // D layout: VGPR r holds M=r (lanes 0-15) / M=r+8 (lanes 16-31)
    float* obase = out + (size_t)b * W_ * F_ + (size_t)wbase * F_ + fbase + n;
#pragma unroll
    for (int r = 0; r < 8; ++r) {
        int m = r + half * 8;
        obase[(size_t)m * F_] = (acc0[r] + acc1[r]) * wscale[m];
    }
}

// ---------------------------------------------------------------------------
// Kernel 3: slot 0 = mean over all S subword positions (sentence embedding).
// Coalesced over f; 512-deep column reduction from L2-resident workspace.
// ---------------------------------------------------------------------------
__global__ void sentence_mean(const float* __restrict__ sub,
                              float* __restrict__ out) {
    int f = blockIdx.x * blockDim.x + threadIdx.x;   // 0..767
    int b = blockIdx.y;
    const float* p = sub + (size_t)b * S_ * F_ + f;
    float acc = 0.f;
#pragma unroll 8
    for (int s = 0; s < S_; ++s) acc += p[(size_t)s * F_];
    out[(size_t)b * W_ * F_ + f] = acc * (1.f / (float)S_);
}

extern "C" void kernel_launch(void* const* d_in, const int* in_sizes, int n_in,
                              void* d_out, int out_size, void* d_ws, size_t ws_size,
                              hipStream_t stream) {
    const float* h     = (const float*)d_in[0];  // (L,B,S,F) f32
    const float* lw    = (const float*)d_in[1];  // (L,) f32
    const float* gamma = (const float*)d_in[2];  // (1,) f32
    const int*   wid   = (const int*)  d_in[3];  // (B,S) i32, sorted per row
    float* out = (float*)d_out;                  // (B, 257, F) f32
    float* sub = (float*)d_ws;                   // needs B*S*F*4 = 48 MiB scratch

    const size_t N = (size_t)B_ * S_ * F_;
    int mix_blocks = (int)((N / 4 + 255) / 256);                 // 12288
    mix_kernel<<<mix_blocks, 256, 0, stream>>>(h, lw, gamma, sub);

    seg_mean_wmma<<<dim3(F_ / 16, 16, B_), 32, 0, stream>>>(sub, wid, out);

    sentence_mean<<<dim3(F_ / 256, B_), 256, 0, stream>>>(sub, out);
}